// LSNNRecurrent_88098369175800
// MI455X (gfx1250) — compile-verified
//
#include <hip/hip_runtime.h>

typedef __attribute__((ext_vector_type(2))) float    v2f;
typedef __attribute__((ext_vector_type(8))) float    v8f;
typedef __attribute__((ext_vector_type(4))) unsigned u32x4;
typedef __attribute__((ext_vector_type(4))) int      i32x4;
typedef __attribute__((ext_vector_type(8))) int      i32x8;

static constexpr int T = 1024, B = 64, I = 256, H = 512;
static constexpr float TAU = 0.5f;   // tau_syn
static constexpr float VTH = 1.0f;   // adaptive threshold b (constant)

// ---- fp32 WMMA 16x16x4: D = A(16x4) * B(4x16) + C ------------------------
__device__ __forceinline__ v8f wmma4f32(v2f a, v2f b, v8f c) {
  return __builtin_amdgcn_wmma_f32_16x16x4_f32(
      /*neg_a=*/false, a, /*neg_b=*/false, b,
      /*c_mod=*/(short)0, c, /*reuse_a=*/false, /*reuse_b=*/false);
}

// ---- Tensor Data Mover: 2D tile global -> LDS -----------------------------
// D# per CDNA5 ISA. 2D tensor (groups 2/3 zero).
// pad_interval is fixed at 7 (pad after every 256 DWORDs stored);
// pad_amount_enc: encoded DWORD count minus 1 (1 -> 2 dwords, 3 -> 4 dwords).
__device__ __forceinline__ void tdm_load_2d(unsigned lds_off, const void* gptr,
                                            unsigned dim0, unsigned dim1,
                                            unsigned stride0,
                                            unsigned pad_amount_enc) {
  const unsigned long long ga = (unsigned long long)(size_t)gptr;
  u32x4 g0;
  g0.x = 0x1u;                                        // count=1, user D#
  g0.y = lds_off;                                     // lds_addr (bytes)
  g0.z = (unsigned)(ga & 0xFFFFFFFFu);                // global_addr[31:0]
  g0.w = (unsigned)((ga >> 32) & 0x01FFFFFFu)         // global_addr[56:32]
         | (2u << 30);                                // type=2 ("image")
  i32x8 g1;
  g1[0] = (int)((2u << 16)                            // data_size = 4B
                | (1u << 20)                          // pad_enable
                | (7u << 22)                          // pad_interval = 256 dw
                | (pad_amount_enc << 25));            // pad_amount
  g1[1] = (int)(dim0 << 16);                          // tensor_dim0[15:0]
  g1[2] = (int)((dim0 >> 16) | (dim1 << 16));         // dim0 hi | dim1 lo
  g1[3] = (int)((dim1 >> 16) | (dim0 << 16));         // dim1 hi | tile_dim0
  g1[4] = (int)(dim1 & 0xFFFFu);                      // tile_dim1 (tile_dim2=0)
  g1[5] = (int)stride0;                               // tensor_dim0_stride lo
  g1[6] = 0;                                          // stride0 hi | stride1 lo
  g1[7] = 0;                                          // stride1 hi
  i32x4 g2 = {0, 0, 0, 0};
  i32x4 g3 = {0, 0, 0, 0};
#if __clang_major__ >= 23
  i32x8 g4 = {0, 0, 0, 0, 0, 0, 0, 0};
  __builtin_amdgcn_tensor_load_to_lds(g0, g1, g2, g3, g4, 0);
#else
  __builtin_amdgcn_tensor_load_to_lds(g0, g1, g2, g3, 0);
#endif
}

// ===========================================================================
// Kernel 1: xin[m][n] = sum_k x[m][k] * Wi[n][k]   (M=T*B, K=I, N=H)
// Wi panel (64 cols x 256 K = 64 KB) shared by all 8 waves, DMA'd into LDS by
// the TDM with 4-DWORD row padding -> stride 260 (bank-conflict-free b64).
// ===========================================================================
#define LWI_STRIDE 260   // 256 + 4 pad floats (TDM pad: 4 dw per 256 dw row)

__global__ void __launch_bounds__(256, 2)
xin_gemm(const float* __restrict__ X, const float* __restrict__ Wi,
         float* __restrict__ Y) {
  const int M = T * B, K = I, N = H;
  const int wave = threadIdx.x >> 5;
  const int lane = threadIdx.x & 31;
  const int l    = lane & 15;
  const int kh   = (lane >> 4) << 1;          // 0 or 2
  const int m0   = blockIdx.x * 128 + wave * 16;
  const int n0   = blockIdx.y * 64;

  __shared__ float lwi[64 * LWI_STRIDE];      // 66.5 KB

  if (threadIdx.x < 32) {                     // wave 0 issues the DMA
    tdm_load_2d((unsigned)(size_t)&lwi[0], Wi + (size_t)n0 * K,
                /*dim0=*/K, /*dim1=*/64, /*stride0=*/K, /*pad_enc=*/3u);
  }
  __builtin_amdgcn_s_wait_tensorcnt(0);
  __syncthreads();

  const float* arow = X + (size_t)(m0 + l) * K + kh;    // A: row m0+l, K pair
  const float* lb   = &lwi[(size_t)l * LWI_STRIDE + kh];

  v8f acc0 = {}, acc1 = {}, acc2 = {}, acc3 = {};
  for (int k0 = 0; k0 < K; k0 += 4) {
    v2f a  = *(const v2f*)(arow + k0);
    v2f b0 = *(const v2f*)(lb + k0);
    v2f b1 = *(const v2f*)(lb + 16 * LWI_STRIDE + k0);
    v2f b2 = *(const v2f*)(lb + 32 * LWI_STRIDE + k0);
    v2f b3 = *(const v2f*)(lb + 48 * LWI_STRIDE + k0);
    acc0 = wmma4f32(a, b0, acc0);
    acc1 = wmma4f32(a, b1, acc1);
    acc2 = wmma4f32(a, b2, acc2);
    acc3 = wmma4f32(a, b3, acc3);
  }

  const int mrow = m0 + ((lane >> 4) << 3);   // +8 for hi half-wave
  float* yb = Y + n0 + l;
#pragma unroll
  for (int j = 0; j < 8; ++j) {
    size_t r = (size_t)(mrow + j) * N;
    yb[r +  0] = acc0[j];
    yb[r + 16] = acc1[j];
    yb[r + 32] = acc2[j];
    yb[r + 48] = acc3[j];
  }
}

// ===========================================================================
// Kernel 2: persistent LSNN scan. 16 blocks x 128 threads = 64 waves.
// Block b owns H columns [b*32, b*32+32). Its 32x512 Wr slice (64 KB) is
// TDM-DMA'd into LDS ONCE and reused for all 1024 steps. TDM pads 2 DWORDs
// after every 256 stored -> LDS row stride 516, second half of each row
// shifted +2 (handled by splitting the K loop).
// ===========================================================================
#define NBLK 16
#define LWR_STRIDE 516   // 512 + 2*2 pad floats per row

__global__ void init_barrier(unsigned* bar) {
  if (threadIdx.x < 2) bar[threadIdx.x] = 0u;
}

__global__ void __launch_bounds__(128, 1)
lsnn_scan(const float* __restrict__ xin, const float* __restrict__ Wr,
          const float* __restrict__ bias, float* __restrict__ out,
          unsigned* __restrict__ bar) {
  const int lane = threadIdx.x & 31;
  const int l    = lane & 15;
  const int kh   = (lane >> 4) << 1;
  const int mt   = threadIdx.x >> 5;          // batch tile 0..3 (wave in block)
  const int n0   = blockIdx.x * 32;           // H group, shared by the block
  const int m0   = mt * 16;
  const int mrow = m0 + ((lane >> 4) << 3);

  __shared__ float lwr[32 * LWR_STRIDE];      // 66 KB: Wr[n0..n0+32) x [0,512)

  if (threadIdx.x < 32) {                     // wave 0 issues the DMA
    tdm_load_2d((unsigned)(size_t)&lwr[0], Wr + (size_t)n0 * H,
                /*dim0=*/H, /*dim1=*/32, /*stride0=*/H, /*pad_enc=*/1u);
  }
  __builtin_amdgcn_s_wait_tensorcnt(0);
  __syncthreads();

  // per-lane state: 16 elements (8 rows x 2 N-tiles), C/D layout
  float vmem[16], cur[16], zp[16];
#pragma unroll
  for (int j = 0; j < 16; ++j) { vmem[j] = 0.f; cur[j] = 0.f; zp[j] = 0.f; }
  const float bz0 = bias[n0 + l];
  const float bz1 = bias[n0 + 16 + l];

  const float* lb0 = &lwr[(size_t)l * LWR_STRIDE + kh];          // B tile 0
  const float* lb1 = &lwr[(size_t)(16 + l) * LWR_STRIDE + kh];   // B tile 1

  for (int t = 0; t < T; ++t) {
    v8f acc0 = {}, acc1 = {};
    if (t > 0) {
      // A frag: z_{t-1}[m0+l][k], read back from the output buffer (L2)
      const float* zrow = out + ((size_t)(t - 1) * B + (m0 + l)) * H + kh;
      for (int k0 = 0; k0 < 256; k0 += 4) {     // first half of each Wr row
        v2f a  = *(const v2f*)(zrow + k0);
        v2f b0 = *(const v2f*)(lb0 + k0);
        v2f b1 = *(const v2f*)(lb1 + k0);
        acc0 = wmma4f32(a, b0, acc0);
        acc1 = wmma4f32(a, b1, acc1);
      }
      for (int k0 = 256; k0 < 512; k0 += 4) {   // second half: +2 pad shift
        v2f a  = *(const v2f*)(zrow + k0);
        v2f b0 = *(const v2f*)(lb0 + k0 + 2);
        v2f b1 = *(const v2f*)(lb1 + k0 + 2);
        acc0 = wmma4f32(a, b0, acc0);
        acc1 = wmma4f32(a, b1, acc1);
      }
    }
    // LIF elementwise update + spike
    const float* xt = xin + (size_t)t * B * H;
    float*       zt = out + (size_t)t * B * H;
#pragma unroll
    for (int j = 0; j < 8; ++j) {
      const size_t r = (size_t)(mrow + j) * H;
      // N-tile 0
      float i0 = TAU * cur[j] + xt[r + n0 + l] + acc0[j];
      float v0 = TAU * vmem[j] * (1.0f - zp[j]) + i0 + bz0;
      float z0 = (v0 - VTH) > 0.0f ? 1.0f : 0.0f;
      cur[j] = i0; vmem[j] = v0; zp[j] = z0;
      zt[r + n0 + l] = z0;
      // N-tile 1
      float i1 = TAU * cur[j + 8] + xt[r + n0 + 16 + l] + acc1[j];
      float v1 = TAU * vmem[j + 8] * (1.0f - zp[j + 8]) + i1 + bz1;
      float z1 = (v1 - VTH) > 0.0f ? 1.0f : 0.0f;
      cur[j + 8] = i1; vmem[j + 8] = v1; zp[j + 8] = z1;
      zt[r + n0 + 16 + l] = z1;
    }
    // ---- grid-wide barrier (two-phase: arrive counter + generation) ----
    __threadfence();
    __syncthreads();
    if (threadIdx.x == 0) {
      unsigned gen = atomicAdd(&bar[1], 0u);
      if (atomicAdd(&bar[0], 1u) == NBLK - 1) {
        atomicExch(&bar[0], 0u);
        __threadfence();
        atomicAdd(&bar[1], 1u);
      } else {
        while (atomicAdd(&bar[1], 0u) == gen) { __builtin_amdgcn_s_sleep(1); }
      }
    }
    __syncthreads();
  }

  // ---- final state tuple: (z, v, i, b) appended after outputs ----
  float* zf  = out + (size_t)T * B * H;
  float* vf  = zf + (size_t)B * H;
  float* if_ = vf + (size_t)B * H;
  float* bf  = if_ + (size_t)B * H;
#pragma unroll
  for (int j = 0; j < 8; ++j) {
    const size_t r = (size_t)(mrow + j) * H;
    zf[r + n0 + l]       = zp[j];
    zf[r + n0 + 16 + l]  = zp[j + 8];
    vf[r + n0 + l]       = vmem[j];
    vf[r + n0 + 16 + l]  = vmem[j + 8];
    if_[r + n0 + l]      = cur[j];
    if_[r + n0 + 16 + l] = cur[j + 8];
    bf[r + n0 + l]       = VTH;
    bf[r + n0 + 16 + l]  = VTH;
  }
}

// ===========================================================================
extern "C" void kernel_launch(void* const* d_in, const int* in_sizes, int n_in,
                              void* d_out, int out_size, void* d_ws, size_t ws_size,
                              hipStream_t stream) {
  const float* x    = (const float*)d_in[0];
  const float* Wi   = (const float*)d_in[1];
  const float* Wr   = (const float*)d_in[2];
  const float* bias = (const float*)d_in[3];
  float* out = (float*)d_out;

  unsigned* bar = (unsigned*)d_ws;                 // barrier state at offset 0
  float* xin = (float*)((char*)d_ws + 256);        // [T,B,H] scratch (128 MB)

  init_barrier<<<1, 32, 0, stream>>>(bar);

  dim3 g1((T * B) / 128, H / 64);                  // 512 x 8 blocks
  xin_gemm<<<g1, 256, 0, stream>>>(x, Wi, xin);

  lsnn_scan<<<NBLK, 128, 0, stream>>>(xin, Wr, bias, out, bar);
}